// MaxPoolingLayer_62895501082689
// MI455X (gfx1250) — compile-verified
//
#include <hip/hip_runtime.h>
#include <hip/hip_bf16.h>

// Per-row argmax-keep: out[i,j] = x[i,j] if j == argmax_j x[i,:], else 0.
// Shape: 16384 rows x 4096 cols, f32. Pure bandwidth problem:
// 256 MiB in + 256 MiB out @ 23.3 TB/s ~= 22 us floor. Strategy: one WG per
// row, async-DMA the row into LDS (gfx1250 async global->LDS path), wave32
// shuffle reduction, non-temporal 128-bit coalesced stores.

typedef float v4f __attribute__((ext_vector_type(4)));
typedef int   v4i_gcc __attribute__((vector_size(16)));   // matches builtin param

#ifndef __has_builtin
#define __has_builtin(x) 0
#endif

#if defined(__gfx1250__) && \
    __has_builtin(__builtin_amdgcn_global_load_async_to_lds_b128) && \
    __has_builtin(__builtin_amdgcn_s_wait_asynccnt)
#define HAVE_ASYNC_LDS 1
#else
#define HAVE_ASYNC_LDS 0
#endif

#define AM_COLS 4096
#define AM_TPB  256
#define AM_VEC  4
#define AM_ITER (AM_COLS / (AM_TPB * AM_VEC))   // 4 float4's per thread
#define AM_WAVES (AM_TPB / 32)

#if HAVE_ASYNC_LDS
typedef __attribute__((address_space(1))) v4i_gcc* gvec_ptr;   // prints as __device__
typedef __attribute__((address_space(3))) v4i_gcc* lvec_ptr;   // prints as __shared__/AS3
#endif

__global__ __launch_bounds__(AM_TPB)
void rowargmax_keep_kernel(const float* __restrict__ x, float* __restrict__ out) {
    __shared__ float lrow[AM_COLS];       // 16 KB staged row
    __shared__ float swv[AM_WAVES];
    __shared__ int   swi[AM_WAVES];
    __shared__ int   swin;

    const int tid = threadIdx.x;
    const size_t rowbase = (size_t)blockIdx.x * AM_COLS;
    const float* xr   = x   + rowbase;
    float*       outr = out + rowbase;

    // ---- Stage row into LDS ----
#if HAVE_ASYNC_LDS
    #pragma unroll
    for (int i = 0; i < AM_ITER; ++i) {
        const int e = (i * AM_TPB + tid) * AM_VEC;   // coalesced float4 index
        __builtin_amdgcn_global_load_async_to_lds_b128(
            (gvec_ptr)(xr + e),
            (lvec_ptr)(lrow + e),
            /*imm offset*/0, /*cpol*/0);
    }
    __builtin_amdgcn_s_wait_asynccnt(0);
#else
    #pragma unroll
    for (int i = 0; i < AM_ITER; ++i) {
        const int e = (i * AM_TPB + tid) * AM_VEC;
        v4f v = __builtin_nontemporal_load((const v4f*)(xr + e));
        *(v4f*)(lrow + e) = v;
    }
#endif
    __syncthreads();

    // ---- Per-thread scan (ascending column order; strict '>' keeps first max) ----
    float best  = -__builtin_inff();
    int   besti = 0;
    #pragma unroll
    for (int i = 0; i < AM_ITER; ++i) {
        const int e = (i * AM_TPB + tid) * AM_VEC;
        v4f v = *(const v4f*)(lrow + e);
        #pragma unroll
        for (int k = 0; k < AM_VEC; ++k) {
            const float vv = v[k];
            const int   cc = e + k;
            if (vv > best) { best = vv; besti = cc; }
        }
    }

    // ---- wave32 butterfly reduction (tie-break: lower column index wins) ----
    #pragma unroll
    for (int off = 16; off >= 1; off >>= 1) {
        const float ov = __shfl_xor(best,  off, 32);
        const int   oi = __shfl_xor(besti, off, 32);
        if (ov > best || (ov == best && oi < besti)) { best = ov; besti = oi; }
    }

    if ((tid & 31) == 0) { swv[tid >> 5] = best; swi[tid >> 5] = besti; }
    __syncthreads();

    if (tid == 0) {
        float b  = swv[0];
        int   bi = swi[0];
        #pragma unroll
        for (int w = 1; w < AM_WAVES; ++w) {
            if (swv[w] > b || (swv[w] == b && swi[w] < bi)) { b = swv[w]; bi = swi[w]; }
        }
        swin = bi;
    }
    __syncthreads();
    const int win = swin;

    // ---- Masked write-back from LDS, non-temporal 128-bit coalesced stores ----
    #pragma unroll
    for (int i = 0; i < AM_ITER; ++i) {
        const int e = (i * AM_TPB + tid) * AM_VEC;
        v4f v = *(const v4f*)(lrow + e);
        v4f o;
        #pragma unroll
        for (int k = 0; k < AM_VEC; ++k) {
            o[k] = ((e + k) == win) ? v[k] : 0.0f;
        }
        __builtin_nontemporal_store(o, (v4f*)(outr + e));
    }
}

extern "C" void kernel_launch(void* const* d_in, const int* in_sizes, int n_in,
                              void* d_out, int out_size, void* d_ws, size_t ws_size,
                              hipStream_t stream) {
    (void)n_in; (void)out_size; (void)d_ws; (void)ws_size;
    const float* x  = (const float*)d_in[0];
    float* out      = (float*)d_out;
    const int rows  = in_sizes[0] / AM_COLS;   // 16384
    rowargmax_keep_kernel<<<rows, AM_TPB, 0, stream>>>(x, out);
}